// GATLayer_33466385170685
// MI455X (gfx1250) — compile-verified
//
#include <hip/hip_runtime.h>
#include <math.h>

typedef __attribute__((ext_vector_type(2))) float v2f;
typedef __attribute__((ext_vector_type(8))) float v8f;
typedef __attribute__((ext_vector_type(4))) int   v4i;

#define GN     4096
#define IN_F   256
#define OUT_F  128
#define NHEAD  2
#define ALPHA  0.2f
#define NEGINF -9000000000000000.0f
#define JT     64    // j-tile width for the flash pass

#define GLOBAL_AS __attribute__((address_space(1)))
#define LDS_AS    __attribute__((address_space(3)))

typedef GLOBAL_AS v4i* g_v4i_ptr;
typedef LDS_AS    v4i* l_v4i_ptr;

#if __has_builtin(__builtin_amdgcn_global_load_async_to_lds_b128)
#define HAVE_ASYNC_LDS 1
#else
#define HAVE_ASYNC_LDS 0
#endif

__device__ __forceinline__ void wait_async_lds() {
#if __has_builtin(__builtin_amdgcn_s_wait_asynccnt)
  __builtin_amdgcn_s_wait_asynccnt(0);
#else
  asm volatile("s_wait_asynccnt 0x0" ::: "memory");
#endif
}

// ---------------------------------------------------------------------------
// Kernel 1: xt[h] = x @ W[h]  (fp32 WMMA 16x16x4),  e1/e2 = xt . a1/a2
// grid = (H, N/16), block = 256 threads (8 waves); wave w -> out cols [16w,16w+16)
// ---------------------------------------------------------------------------
__global__ __launch_bounds__(256) void gat_proj(const float* __restrict__ x,
                                                const float* __restrict__ W,
                                                const float* __restrict__ a,
                                                float* __restrict__ xt,
                                                float* __restrict__ e1g,
                                                float* __restrict__ e2g) {
  const int h    = blockIdx.x;
  const int rb   = blockIdx.y;
  const int tid  = threadIdx.x;
  const int lane = tid & 31;
  const int wave = tid >> 5;
  const int l16  = lane & 15;
  const int lh   = lane >> 4;      // 0: lanes 0-15, 1: lanes 16-31
  const int row0 = rb * 16;
  const int col0 = wave * 16;

  const float* Wh   = W + (size_t)h * IN_F * OUT_F;
  const float* xrow = x + (size_t)(row0 + l16) * IN_F;

  v8f c = {};
  for (int k0 = 0; k0 < IN_F; k0 += 4) {
    const int k = k0 + 2 * lh;
    v2f av, bv;
    av.x = xrow[k];                       // A: M=l16, K=k
    av.y = xrow[k + 1];
    bv.x = Wh[(size_t)k * OUT_F + col0 + l16];       // B: K=k, N=l16
    bv.y = Wh[(size_t)(k + 1) * OUT_F + col0 + l16];
    c = __builtin_amdgcn_wmma_f32_16x16x4_f32(false, av, false, bv, (short)0, c,
                                              false, false);
  }

  __shared__ float e1s[16];
  __shared__ float e2s[16];
  if (tid < 16) { e1s[tid] = 0.f; e2s[tid] = 0.f; }
  __syncthreads();

  const float a1v = a[h * (2 * OUT_F) + col0 + l16];
  const float a2v = a[h * (2 * OUT_F) + OUT_F + col0 + l16];
  float* xth = xt + (size_t)h * GN * OUT_F;
#pragma unroll
  for (int r = 0; r < 8; ++r) {
    const int m = r + 8 * lh;            // C layout: row = r + 8*(lane>=16)
    const float v = c[r];
    xth[(size_t)(row0 + m) * OUT_F + col0 + l16] = v;
    atomicAdd(&e1s[m], v * a1v);         // ds_add_f32
    atomicAdd(&e2s[m], v * a2v);
  }
  __syncthreads();
  if (tid < 16) {
    e1g[h * GN + row0 + tid] = e1s[tid];
    e2g[h * GN + row0 + tid] = e2s[tid];
  }
}

// ---------------------------------------------------------------------------
// Kernel 2: fused masked-softmax attention + attn@xt + ELU (flash style)
// grid = (H, N/16), block = 256 threads (8 waves)
// ---------------------------------------------------------------------------
__global__ __launch_bounds__(256) void gat_attn(const float* __restrict__ adj,
                                                const float* __restrict__ xt,
                                                const float* __restrict__ e1g,
                                                const float* __restrict__ e2g,
                                                float* __restrict__ out) {
  const int h    = blockIdx.x;
  const int rb   = blockIdx.y;
  const int tid  = threadIdx.x;
  const int lane = tid & 31;
  const int wave = tid >> 5;
  const int l16  = lane & 15;
  const int lh   = lane >> 4;
  const int row0 = rb * 16;
  const int col0 = wave * 16;

  __shared__ float sT[16][JT + 2];          // score / P tile (padded: banks)
  __shared__ float xtT[JT][OUT_F + 4];      // staged xt tile (row = 528B, 16B aligned)
  __shared__ float red[16][16];             // row-reduction partials
  __shared__ float mrow[16], lrow[16], srow[16], e1s[16];

  if (tid < 16) {
    mrow[tid] = -INFINITY;
    lrow[tid] = 0.f;
    e1s[tid]  = e1g[h * GN + row0 + tid];
  }
  v8f acc = {};
  __syncthreads();

  const float* xth = xt + (size_t)h * GN * OUT_F;
  const float* e2h = e2g + h * GN;

  const int r  = tid >> 4;                  // tile row this thread scores
  const int cs = (tid & 15) * (JT / 16);    // 4-col segment (16B aligned)

  for (int j0 = 0; j0 < GN; j0 += JT) {
    // -- stage xt tile (JT x 128): async direct-to-LDS if available --
#pragma unroll
    for (int it = 0; it < (JT * OUT_F) / (256 * 4); ++it) {
      const int q  = tid + 256 * it;
      const int rr = q >> 5;
      const int c4 = (q & 31) * 4;
      const float* gsrc = xth + (size_t)(j0 + rr) * OUT_F + c4;
#if HAVE_ASYNC_LDS
      __builtin_amdgcn_global_load_async_to_lds_b128(
          (g_v4i_ptr)(void*)const_cast<float*>(gsrc),
          (l_v4i_ptr)(void*)&xtT[rr][c4], 0, 0);
#else
      const float4 v = *(const float4*)gsrc;
      xtT[rr][c4 + 0] = v.x;
      xtT[rr][c4 + 1] = v.y;
      xtT[rr][c4 + 2] = v.z;
      xtT[rr][c4 + 3] = v.w;
#endif
    }

    // -- masked leaky-relu scores (vectorized adj/e2 reads) --
    const int irow = row0 + r;
    const float4 adj4 = *(const float4*)(adj + (size_t)irow * GN + j0 + cs);
    const float4 e24  = *(const float4*)(e2h + j0 + cs);
    float pm = -INFINITY;
    float sv[JT / 16];
    const float adjv[4] = {adj4.x, adj4.y, adj4.z, adj4.w};
    const float e2v[4]  = {e24.x, e24.y, e24.z, e24.w};
#pragma unroll
    for (int u = 0; u < JT / 16; ++u) {
      const int j = j0 + cs + u;
      const bool valid = (adjv[u] > 0.f) || (j == irow);   // adj + I
      const float sc = e1s[r] + e2v[u];
      const float lr = sc > 0.f ? sc : ALPHA * sc;
      const float s  = valid ? lr : NEGINF;
      sv[u] = s;
      sT[r][cs + u] = s;
      pm = fmaxf(pm, s);
    }
    red[r][tid & 15] = pm;

    if (j0 + JT < GN)                       // gfx1250 global_prefetch_b8
      __builtin_prefetch(&adj[(size_t)irow * GN + j0 + JT + cs], 0, 1);
#if HAVE_ASYNC_LDS
    wait_async_lds();
#endif
    __syncthreads();

    // -- online row max + rescale factor --
    if (tid < 16) {
      float mn = mrow[tid];
#pragma unroll
      for (int g = 0; g < 16; ++g) mn = fmaxf(mn, red[tid][g]);
      srow[tid] = __expf(mrow[tid] - mn);
      mrow[tid] = mn;
    }
    __syncthreads();

    // -- P = exp(s - m), partial row sums --
    float ps = 0.f;
    const float mr = mrow[r];
#pragma unroll
    for (int u = 0; u < JT / 16; ++u) {
      const float p = __expf(sv[u] - mr);
      sT[r][cs + u] = p;
      ps += p;
    }
    red[r][tid & 15] = ps;
    __syncthreads();

    if (tid < 16) {
      float t = 0.f;
#pragma unroll
      for (int g = 0; g < 16; ++g) t += red[tid][g];
      lrow[tid] = lrow[tid] * srow[tid] + t;
    }

    // -- rescale accumulator, then acc += P(16xJT) @ xtT(JT x 16cols) --
#pragma unroll
    for (int q = 0; q < 8; ++q) acc[q] *= srow[q + 8 * lh];

    for (int k0 = 0; k0 < JT; k0 += 4) {
      const int k = k0 + 2 * lh;
      v2f av, bv;
      av.x = sT[l16][k];                    // A: M=l16, K=k
      av.y = sT[l16][k + 1];
      bv.x = xtT[k][col0 + l16];            // B: K=k, N=l16
      bv.y = xtT[k + 1][col0 + l16];
      acc = __builtin_amdgcn_wmma_f32_16x16x4_f32(false, av, false, bv,
                                                  (short)0, acc, false, false);
    }
    __syncthreads();
  }

  // -- normalize, ELU, store: out[i, h*128 + o] --
#pragma unroll
  for (int q = 0; q < 8; ++q) {
    const int m = q + 8 * lh;
    const float v = acc[q] / lrow[m];
    const float o = v > 0.f ? v : (__expf(v) - 1.f);
    out[(size_t)(row0 + m) * (NHEAD * OUT_F) + h * OUT_F + col0 + l16] = o;
  }
}

// ---------------------------------------------------------------------------
extern "C" void kernel_launch(void* const* d_in, const int* in_sizes, int n_in,
                              void* d_out, int out_size, void* d_ws, size_t ws_size,
                              hipStream_t stream) {
  const float* x   = (const float*)d_in[0];   // (N, IN_F)
  const float* adj = (const float*)d_in[1];   // (N, N)
  const float* W   = (const float*)d_in[2];   // (H, IN_F, OUT_F)
  const float* a   = (const float*)d_in[3];   // (H, 2*OUT_F, 1)
  float* out = (float*)d_out;                 // (N, H*OUT_F)

  float* xtws = (float*)d_ws;                       // H*N*OUT_F = 4 MB
  float* e1   = xtws + (size_t)NHEAD * GN * OUT_F;  // H*N
  float* e2   = e1 + (size_t)NHEAD * GN;            // H*N

  dim3 grid(NHEAD, GN / 16);
  gat_proj<<<grid, 256, 0, stream>>>(x, W, a, xtws, e1, e2);
  gat_attn<<<grid, 256, 0, stream>>>(adj, xtws, e1, e2, out);
}